// VQVAEQuantizer_19258633355614
// MI455X (gfx1250) — compile-verified
//
#include <hip/hip_runtime.h>
#include <hip/hip_bf16.h>
#include <math.h>

// ---------------------------------------------------------------------------
// VQ-VAE quantizer for MI455X (gfx1250, wave32).
// Shapes: B=16, C=512, H=W=32  -> P = B*H*W = 16384 positions
//         N = 16384 codes, D = 32
// Pipeline:
//   1) e_norm = normalize(emb_weight)                       [N,32]
//   2) z = normalize(enc @ proj_w^T + proj_b)  (WMMA f32, proj_w in LDS)
//   3) fused sims GEMM (P x N x 32) + argmax   (WMMA f32, e chunks double-
//      buffered in LDS via async global->LDS)  -> idx[P]
//   4) gather latents, MSE partials, int histogram
//   5) out = latents @ exp_w^T + exp_b         (WMMA f32, exp_w in LDS)
//   6) finalize loss + perplexity (deterministic ordered reductions)
// ---------------------------------------------------------------------------

typedef __attribute__((ext_vector_type(2))) float v2f;
typedef __attribute__((ext_vector_type(8))) float v8f;
typedef __attribute__((ext_vector_type(4))) int   v4i;

#define NPOS   16384   // B*H*W
#define NCODE  16384
#define DDIM   32
#define CDIM   512
#define HWSZ   1024    // H*W
#define OUTEL  8388608 // 16*512*32*32

#define CHUNK_CODES  128                    // codes staged per LDS chunk
#define CHUNK_FLOATS (CHUNK_CODES * DDIM)   // 4096 floats = 16 KB

// ---- async global->LDS copy (CDNA5), with portable fallback ----------------
#if defined(__has_builtin)
#if __has_builtin(__builtin_amdgcn_global_load_async_to_lds_b128)
#define HAVE_ASYNC_LDS 1
#endif
#endif
#ifndef HAVE_ASYNC_LDS
#define HAVE_ASYNC_LDS 0
#endif

#if HAVE_ASYNC_LDS
typedef __attribute__((address_space(1))) v4i as1_v4i;
typedef __attribute__((address_space(3))) v4i as3_v4i;
#endif

__device__ __forceinline__ void async_copy_b128(const float* gsrc, float* ldst) {
#if HAVE_ASYNC_LDS
    __builtin_amdgcn_global_load_async_to_lds_b128(
        (as1_v4i*)gsrc, (as3_v4i*)ldst, 0, 0);
#else
    *(float4*)ldst = *(const float4*)gsrc;
#endif
}

__device__ __forceinline__ void async_wait_all() {
#if HAVE_ASYNC_LDS
#if __has_builtin(__builtin_amdgcn_s_wait_asynccnt)
    __builtin_amdgcn_s_wait_asynccnt(0);
#else
    asm volatile("s_wait_asynccnt 0x0" ::: "memory");
#endif
#endif
}

// --- 1) normalize embedding rows: one wave per row (lane = d) ---------------
__global__ void k_norm_emb(const float* __restrict__ w, float* __restrict__ e) {
    int wave  = (blockIdx.x * blockDim.x + threadIdx.x) >> 5;
    int lane  = threadIdx.x & 31;
    int nwav  = (gridDim.x * blockDim.x) >> 5;
    for (int n = wave; n < NCODE; n += nwav) {
        float v = w[(size_t)n * DDIM + lane];
        float s = v * v;
        s += __shfl_xor(s, 16, 32);
        s += __shfl_xor(s, 8, 32);
        s += __shfl_xor(s, 4, 32);
        s += __shfl_xor(s, 2, 32);
        s += __shfl_xor(s, 1, 32);
        e[(size_t)n * DDIM + lane] = v / fmaxf(sqrtf(s), 1e-6f);
    }
}

// --- 2) projection GEMM + bias + L2-normalize -------------------------------
//  A = enc (M=pos, K=c), B(k,n) = proj_w[n*512 + k]; proj_w (64 KB) in LDS.
__global__ void __launch_bounds__(256) k_project(const float* __restrict__ enc,
                          const float* __restrict__ pw,
                          const float* __restrict__ pb,
                          float* __restrict__ z) {
    __shared__ float wbuf[DDIM * CDIM];   // 64 KB
#pragma unroll
    for (int q = 0; q < 16; ++q) {
        int off = (q * 256 + threadIdx.x) * 4;
        async_copy_b128(pw + off, wbuf + off);
    }

    int wave = threadIdx.x >> 5;
    int lane = threadIdx.x & 31;
    int m    = lane & 15;
    int hi   = lane >> 4;
    int tile = blockIdx.x * 8 + wave;   // 0..1023
    int p0   = tile * 16;
    int b    = p0 >> 10;
    int hw   = (p0 & 1023) + m;
    const float* encb = enc + (size_t)b * CDIM * HWSZ + hw;

    async_wait_all();
    __syncthreads();

    v8f acc0 = {};
    v8f acc1 = {};
    for (int kk = 0; kk < CDIM; kk += 4) {
        int k0 = kk + 2 * hi;
        v2f a;
        a.x = encb[(size_t)k0 * HWSZ];
        a.y = encb[(size_t)(k0 + 1) * HWSZ];
        v2f b0 = *(const v2f*)(wbuf + (size_t)m * CDIM + k0);
        v2f b1 = *(const v2f*)(wbuf + (size_t)(16 + m) * CDIM + k0);
        acc0 = __builtin_amdgcn_wmma_f32_16x16x4_f32(false, a, false, b0,
                                                     (short)0, acc0, false, false);
        acc1 = __builtin_amdgcn_wmma_f32_16x16x4_f32(false, a, false, b1,
                                                     (short)0, acc1, false, false);
    }
    float bias0 = pb[m];
    float bias1 = pb[16 + m];
#pragma unroll
    for (int r = 0; r < 8; ++r) {
        float v0 = acc0[r] + bias0;
        float v1 = acc1[r] + bias1;
        float s = v0 * v0 + v1 * v1;     // sum over 32 d within 16-lane half
        s += __shfl_xor(s, 1, 32);
        s += __shfl_xor(s, 2, 32);
        s += __shfl_xor(s, 4, 32);
        s += __shfl_xor(s, 8, 32);
        float inv = 1.0f / fmaxf(sqrtf(s), 1e-6f);
        int p = p0 + r + 8 * hi;
        z[(size_t)p * DDIM + m]      = v0 * inv;
        z[(size_t)p * DDIM + 16 + m] = v1 * inv;
    }
}

// --- 3) fused sims GEMM + argmax --------------------------------------------
//  8 waves/block share double-buffered 16 KB chunks of e_norm (async->LDS).
__device__ __forceinline__ void issue_chunk(const float* __restrict__ e,
                                            float* buf, int ch, int tid) {
    const float* src = e + (size_t)ch * CHUNK_FLOATS;
#pragma unroll
    for (int q = 0; q < 4; ++q) {
        int off = (q * 256 + tid) * 4;
        async_copy_b128(src + off, buf + off);
    }
}

__global__ void __launch_bounds__(256) k_argmax(const float* __restrict__ z,
                         const float* __restrict__ e,
                         int* __restrict__ idx) {
    __shared__ float ebuf[2][CHUNK_FLOATS];   // 2 x 16 KB
    int wave = threadIdx.x >> 5;
    int lane = threadIdx.x & 31;
    int m    = lane & 15;
    int hi   = lane >> 4;
    int tile = blockIdx.x * 8 + wave;   // 0..1023
    int p0   = tile * 16;

    // preload A fragments (z rows) for all 8 K-steps: K = 4t + 2*hi + {0,1}
    v2f a[8];
    const float* zr = z + (size_t)(p0 + m) * DDIM + 2 * hi;
#pragma unroll
    for (int t = 0; t < 8; ++t) a[t] = *(const v2f*)(zr + t * 4);

    float maxv[8];
    int   maxi[8];
#pragma unroll
    for (int r = 0; r < 8; ++r) { maxv[r] = -3.0e38f; maxi[r] = 0; }

    issue_chunk(e, ebuf[0], 0, threadIdx.x);
    const int NCHUNK = NCODE / CHUNK_CODES;   // 128
    for (int ch = 0; ch < NCHUNK; ++ch) {
        async_wait_all();
        __syncthreads();                       // chunk ch fully in LDS
        if (ch + 1 < NCHUNK)
            issue_chunk(e, ebuf[(ch + 1) & 1], ch + 1, threadIdx.x);

        const float* ebase = ebuf[ch & 1];
        for (int lt = 0; lt < CHUNK_CODES / 16; ++lt) {
            const float* er = ebase + (size_t)(lt * 16 + m) * DDIM + 2 * hi;
            v8f c = {};
#pragma unroll
            for (int t = 0; t < 8; ++t) {
                v2f bf = *(const v2f*)(er + t * 4);
                c = __builtin_amdgcn_wmma_f32_16x16x4_f32(false, a[t], false, bf,
                                                          (short)0, c, false, false);
            }
            int n = ch * CHUNK_CODES + lt * 16 + m;
#pragma unroll
            for (int r = 0; r < 8; ++r) {
                if (c[r] > maxv[r]) { maxv[r] = c[r]; maxi[r] = n; }  // strict >
            }
        }
        __syncthreads();                       // everyone done reading chunk ch
    }

#pragma unroll
    for (int r = 0; r < 8; ++r) {
        float v = maxv[r];
        int   i = maxi[r];
#pragma unroll
        for (int off = 1; off < 16; off <<= 1) {
            float ov = __shfl_xor(v, off, 32);
            int   oi = __shfl_xor(i, off, 32);
            if (ov > v || (ov == v && oi < i)) { v = ov; i = oi; }
        }
        if (m == 0) idx[p0 + r + 8 * hi] = i;
    }
}

// --- 4) gather latents + loss partials + histogram --------------------------
__global__ void k_gather(const float* __restrict__ z,
                         const float* __restrict__ e,
                         const int* __restrict__ idx,
                         float* __restrict__ lat,
                         int* __restrict__ hist,
                         float* __restrict__ partial) {
    int p  = blockIdx.x * blockDim.x + threadIdx.x;  // 64 blocks x 256 = 16384
    int id = idx[p];
    atomicAdd(&hist[id], 1);  // integer atomic: deterministic
    const float* er = e + (size_t)id * DDIM;
    const float* zr = z + (size_t)p * DDIM;
    float*       lr = lat + (size_t)p * DDIM;
    float s = 0.f;
#pragma unroll
    for (int d = 0; d < DDIM; ++d) {
        float ev = er[d];
        lr[d]    = ev;
        float df = zr[d] - ev;
        s += df * df;
    }
    __shared__ float red[256];
    red[threadIdx.x] = s;
    __syncthreads();
    for (int st = 128; st; st >>= 1) {
        if (threadIdx.x < (unsigned)st) red[threadIdx.x] += red[threadIdx.x + st];
        __syncthreads();
    }
    if (threadIdx.x == 0) partial[blockIdx.x] = red[0];
}

// --- 5) expansion GEMM: exp_w (64 KB) staged in LDS, 32 c-tiles per wave ----
__global__ void __launch_bounds__(256) k_expand(const float* __restrict__ lat,
                         const float* __restrict__ ew,
                         const float* __restrict__ eb,
                         float* __restrict__ out) {
    __shared__ float wbuf[CDIM * DDIM];   // 64 KB
#pragma unroll
    for (int q = 0; q < 16; ++q) {
        int off = (q * 256 + threadIdx.x) * 4;
        async_copy_b128(ew + off, wbuf + off);
    }

    int wave = threadIdx.x >> 5;
    int lane = threadIdx.x & 31;
    int m    = lane & 15;
    int hi   = lane >> 4;
    int tile = blockIdx.x * 8 + wave;   // 0..1023
    int p0   = tile * 16;
    int b    = p0 >> 10;
    int hwb  = p0 & 1023;

    v2f a[8];
    const float* lr = lat + (size_t)(p0 + m) * DDIM + 2 * hi;
#pragma unroll
    for (int t = 0; t < 8; ++t) a[t] = *(const v2f*)(lr + t * 4);

    async_wait_all();
    __syncthreads();

    for (int ct = 0; ct < CDIM / 16; ++ct) {
        int c0 = ct * 16 + m;
        const float* er = wbuf + (size_t)c0 * DDIM + 2 * hi;
        v8f acc = {};
#pragma unroll
        for (int t = 0; t < 8; ++t) {
            v2f bf = *(const v2f*)(er + t * 4);
            acc = __builtin_amdgcn_wmma_f32_16x16x4_f32(false, a[t], false, bf,
                                                        (short)0, acc, false, false);
        }
        float bias = eb[c0];
#pragma unroll
        for (int r = 0; r < 8; ++r) {
            out[(size_t)b * (CDIM * HWSZ) + (size_t)c0 * HWSZ + hwb + r + 8 * hi] =
                acc[r] + bias;
        }
    }
}

// --- 6) finalize: loss (ordered sum of 64 partials) + perplexity ------------
__global__ void k_final(const float* __restrict__ partial,
                        const int* __restrict__ hist,
                        float* __restrict__ scalars) {
    __shared__ float red[256];
    float local = 0.f;
    for (int bin = threadIdx.x; bin < NCODE; bin += 256) {
        float u = (float)hist[bin] * (1.0f / (float)NPOS);
        local += -u * logf(u + 1e-6f);
    }
    red[threadIdx.x] = local;
    __syncthreads();
    for (int st = 128; st; st >>= 1) {
        if (threadIdx.x < (unsigned)st) red[threadIdx.x] += red[threadIdx.x + st];
        __syncthreads();
    }
    if (threadIdx.x == 0) {
        float loss = 0.f;
        for (int i = 0; i < 64; ++i) loss += partial[i];  // fixed order
        scalars[0] = loss * (1.0f / (float)(NPOS * DDIM));
        scalars[1] = expf(red[0]);
    }
}

extern "C" void kernel_launch(void* const* d_in, const int* in_sizes, int n_in,
                              void* d_out, int out_size, void* d_ws, size_t ws_size,
                              hipStream_t stream) {
    const float* enc  = (const float*)d_in[0];
    const float* embw = (const float*)d_in[1];
    const float* pw   = (const float*)d_in[2];
    const float* pb   = (const float*)d_in[3];
    const float* ew   = (const float*)d_in[4];
    const float* eb   = (const float*)d_in[5];
    float* out = (float*)d_out;

    char* ws = (char*)d_ws;
    float* z       = (float*)(ws);                                  // 2 MB
    float* e       = (float*)(ws + (2u << 20));                     // 2 MB
    float* lat     = (float*)(ws + (4u << 20));                     // 2 MB
    int*   idx     = (int*)  (ws + (6u << 20));                     // 64 KB
    int*   hist    = (int*)  (ws + (6u << 20) + (64u << 10));       // 64 KB
    float* partial = (float*)(ws + (6u << 20) + (128u << 10));      // 256 B

    (void)hipMemsetAsync(hist, 0, NCODE * sizeof(int), stream);

    k_norm_emb<<<64, 256, 0, stream>>>(embw, e);
    k_project<<<128, 256, 0, stream>>>(enc, pw, pb, z);     // 1024 waves
    k_argmax<<<128, 256, 0, stream>>>(z, e, idx);           // 1024 waves
    k_gather<<<64, 256, 0, stream>>>(z, e, idx, lat, hist, partial);
    k_expand<<<128, 256, 0, stream>>>(lat, ew, eb, out);    // 1024 waves
    k_final<<<1, 256, 0, stream>>>(partial, hist, out + OUTEL);
}